// Model_53249004536250
// MI455X (gfx1250) — compile-verified
//
#include <hip/hip_runtime.h>
#include <cstdio>

// ---------------------------------------------------------------------------
// Problem constants (match reference)
// ---------------------------------------------------------------------------
#define BB   64          // batch
#define TT   256         // time steps
#define INF  256         // input features (x has IN+1 cols, col0 = cluster id)
#define HH   512         // hidden
#define OO   256         // output
#define CC   16          // clusters
#define BT   (BB*TT)     // 16384 rows
#define KB_H (HH/4)      // 128 K-blocks of 4 in the recurrence

typedef float v2f __attribute__((ext_vector_type(2)));
typedef float v8f __attribute__((ext_vector_type(8)));

// fp32 WMMA: D(16x16,f32) = A(16x4,f32) x B(4x16,f32) + C
// A frag: lane<16 -> row M=lane, K = k0+{0,1}; lane>=16 -> row M=lane-16, K = k0+{2,3}
// B frag: lane<16 -> col N=lane, K = k0+{0,1}; lane>=16 -> col N=lane-16, K = k0+{2,3}
static __device__ __forceinline__ v8f wmma_f32(v2f a, v2f b, v8f c) {
  return __builtin_amdgcn_wmma_f32_16x16x4_f32(false, a, false, b, (short)0, c,
                                               false, false);
}

// ---------------------------------------------------------------------------
// Generic tiled WMMA GEMM:  C[m,n] = relu?( sum_k A[row(m), a_off+k]*B[n,k] + bias[n] )
//   A row-major with leading dim lda, optional row gather (row_idx),
//   B row-major (N x K) with optional per-group matrix select (group_sel),
//   grid = (N/64, Mpg/64, groups); block = 512 (16 waves, 4x4 tile grid).
//   permute_bt: store output row m=(b*TT+t) at row (t*BB+b) instead, giving the
//   recurrence a contiguous 128 KB slab per timestep.
// ---------------------------------------------------------------------------
template <bool RELU>
__global__ __launch_bounds__(512)
void gemm_wmma_kernel(const float* __restrict__ A, int lda, int a_off,
                      const int* __restrict__ row_idx,
                      const float* __restrict__ Bmat,
                      const float* __restrict__ bias,
                      const int* __restrict__ group_sel,
                      int Mpg, int N, int K,
                      float* __restrict__ Cmat, int ldc, int permute_bt) {
  const int tid = threadIdx.x;
  const int n0  = blockIdx.x * 64;
  const int g   = blockIdx.z;
  const int m0  = g * Mpg + blockIdx.y * 64;

  const float* Bp    = Bmat;
  const float* biasp = bias;
  if (group_sel) {
    const int s = group_sel[g];
    Bp = Bmat + (size_t)s * (size_t)N * (size_t)K;
    if (bias) biasp = bias + (size_t)s * (size_t)N;
  }

  // LDS tiles, padded stride 36 floats -> conflict-free 16-row b64 reads
  __shared__ float As[64 * 36];
  __shared__ float Bs[64 * 36];

  const int wave  = tid >> 5;
  const int lane  = tid & 31;
  const int tm    = wave >> 2;          // 0..3 : M tile inside block
  const int tn    = wave & 3;           // 0..3 : N tile inside block
  const int l15   = lane & 15;
  const int lhalf = (lane < 16) ? 0 : 2;

  v8f acc = {};

  // cooperative tile loader: thread -> 4 consecutive floats of one row
  const int lr = tid >> 3;              // row 0..63
  const int lk = (tid & 7) * 4;         // col 0,4,...,28
  int arow = m0 + lr;
  if (row_idx) arow = row_idx[arow];
  const float* aptr = A + (size_t)arow * (size_t)lda + a_off + lk;
  const float* bptr = Bp + (size_t)(n0 + lr) * (size_t)K + lk;

  float ar[4], br[4];
#pragma unroll
  for (int j = 0; j < 4; ++j) { ar[j] = aptr[j]; br[j] = bptr[j]; }

  const float* asm_ = &As[(tm * 16 + l15) * 36 + lhalf];
  const float* bsm_ = &Bs[(tn * 16 + l15) * 36 + lhalf];

  for (int k0 = 0; k0 < K; k0 += 32) {
#pragma unroll
    for (int j = 0; j < 4; ++j) {
      As[lr * 36 + lk + j] = ar[j];
      Bs[lr * 36 + lk + j] = br[j];
    }
    __syncthreads();

    // issue next tile's global loads; they retire while we do WMMAs
    if (k0 + 32 < K) {
#pragma unroll
      for (int j = 0; j < 4; ++j) {
        ar[j] = aptr[k0 + 32 + j];
        br[j] = bptr[k0 + 32 + j];
      }
    }

    v2f afr[8], bfr[8];
#pragma unroll
    for (int s = 0; s < 8; ++s) {       // batch all 16 LDS frag loads
      afr[s] = *(const v2f*)(asm_ + 4 * s);
      bfr[s] = *(const v2f*)(bsm_ + 4 * s);
    }
#pragma unroll
    for (int s = 0; s < 8; ++s) acc = wmma_f32(afr[s], bfr[s], acc);

    __syncthreads();
  }

  // epilogue: C/D layout -> lane holds N=l15, VGPR r holds M = r (+8 for hi half)
  const int   n  = n0 + tn * 16 + l15;
  const float bv = biasp ? biasp[n] : 0.0f;
  const int   mb = (lane < 16) ? 0 : 8;
#pragma unroll
  for (int r = 0; r < 8; ++r) {
    const int m = m0 + tm * 16 + mb + r;
    int mrow = m;
    if (permute_bt) mrow = (m & (TT - 1)) * BB + (m >> 8);  // (b,t) -> (t,b)
    float v = acc[r] + bv;
    if (RELU) v = fmaxf(v, 0.0f);
    Cmat[(size_t)mrow * (size_t)ldc + n] = v;
  }
}

// ---------------------------------------------------------------------------
// Pack Whh (H x H row-major) into fragment-major order so the recurrence's
// B-fragment loads are fully coalesced: element i = (kb*32 + w)*32 + l is the
// float2 consumed by wave w, lane l at K-block kb:
//   n = w*16 + (l&15), k = 4*kb + (l<16 ? 0 : 2)  ->  (Whh[n][k], Whh[n][k+1])
// One wave then streams 32 consecutive float2s (256 B) per K-block, and all
// 128 K-block addresses fit the 24-bit immediate offset of one base register.
// ---------------------------------------------------------------------------
__global__ void pack_whh_kernel(const float* __restrict__ whh,
                                float* __restrict__ wp) {
  const int i = blockIdx.x * 256 + threadIdx.x;   // 0 .. 131071
  const int l  = i & 31;
  const int w  = (i >> 5) & 31;
  const int kb = i >> 10;
  const int n  = w * 16 + (l & 15);
  const int k  = 4 * kb + ((l < 16) ? 0 : 2);
  v2f v;
  v.x = whh[(size_t)n * HH + k];
  v.y = whh[(size_t)n * HH + k + 1];
  ((v2f*)wp)[i] = v;
}

// ---------------------------------------------------------------------------
// Bidirectional RNN recurrence for one layer. Gates G (stored (t,b,h): one
// contiguous 128 KB slab per timestep) are precomputed. One 1024-thread block
// per direction; hidden state h (64x512) lives in LDS (padded stride 516 ->
// conflict-free). Whh streams from the packed fragment-major buffer every
// step (L2-resident). A laundered integer offset (always 0, opaque to the
// compiler) defeats cross-timestep hoisting of the 128 t-invariant fragment
// loads (which caused register spills) WITHOUT destroying pointer provenance,
// so the loads stay global_load_b64 (LOADcnt only) instead of flat_load_b64
// (LOADcnt+DScnt, LDS-path arbitration).
// 2-stage software pipeline: K-block kb+1 is in flight while kb's WMMAs run.
//   hn = relu(G[t,:,:] + h @ Whh^T);  h = mask ? hn : h
//   forward : out[b, t,       dir*512 + n] = mask ? hn : 0
//   backward: out[b, seq-1-t, dir*512 + n] = mask ? hn : 0   (out pre-zeroed)
// ---------------------------------------------------------------------------
#define HSTRIDE 516
#define RNN_LDS_BYTES (BB * HSTRIDE * 4)

__global__ __launch_bounds__(1024)
void rnn_bidir_kernel(const float* __restrict__ Gf, const float* __restrict__ Gb,
                      const float* __restrict__ Wpf, const float* __restrict__ Wpb,
                      const float* __restrict__ h0layer,   // (2, B, H) for this layer
                      const int* __restrict__ seq_len,
                      float* __restrict__ outbuf) {        // (BT, 2H), pre-zeroed
  extern __shared__ float hs[];
  __shared__ int sseq[BB];

  const int    dir = blockIdx.x;
  const float* G   = dir ? Gb : Gf;
  const float* Wp  = dir ? Wpb : Wpf;
  const float* h0p = h0layer + (size_t)dir * BB * HH;
  const int    col0 = dir * HH;

  const int tid = threadIdx.x;
  for (int i = tid; i < BB * HH; i += 1024) {
    const int b = i >> 9, c = i & (HH - 1);
    hs[b * HSTRIDE + c] = h0p[i];
  }
  if (tid < BB) sseq[tid] = seq_len[tid];
  __syncthreads();

  const int wave  = tid >> 5;
  const int lane  = tid & 31;
  const int n0    = wave * 16;                 // 32 waves x 16 cols = 512 cols
  const int l15   = lane & 15;
  const int lhalf = (lane < 16) ? 0 : 2;
  const int mb    = (lane < 16) ? 0 : 8;
  const int n     = n0 + l15;

  // per-mi LDS fragment base addresses
  const float* hb[4];
#pragma unroll
  for (int mi = 0; mi < 4; ++mi)
    hb[mi] = &hs[(mi * 16 + l15) * HSTRIDE + lhalf];

  unsigned toff = 0;  // always 0, but opaque to the compiler (see below)

  for (int t = 0; t < TT; ++t) {
    if (t + 1 < TT) {
      // 1024 threads x 128 B = exactly the next timestep's 128 KB gate slab
      __builtin_prefetch(&G[((size_t)(t + 1) * BB) * HH + tid * 32], 0, 1);
    }

    // launder a zero offset each step: the fragment-load addresses become
    // loop-variant as far as the compiler can prove, so the 128 loads stay
    // inside the t-loop (streamed from L2) instead of hoisted + spilled;
    // Wp's provenance is preserved -> global_load_b64, not flat_load_b64.
    asm volatile("" : "+v"(toff));
    const v2f* wfrag = (const v2f*)Wp + toff + (size_t)wave * 32 + lane;

    v8f acc[4] = {};

    // ---- software-pipelined K loop: load kb+1 while computing kb ----
    v2f bf = wfrag[0];
    v2f a[4];
#pragma unroll
    for (int mi = 0; mi < 4; ++mi) a[mi] = *(const v2f*)(hb[mi]);

#pragma unroll 4
    for (int kb = 0; kb < KB_H - 1; ++kb) {
      v2f bfn = wfrag[(size_t)(kb + 1) * 1024];
      v2f an[4];
#pragma unroll
      for (int mi = 0; mi < 4; ++mi)
        an[mi] = *(const v2f*)(hb[mi] + (kb + 1) * 4);
#pragma unroll
      for (int mi = 0; mi < 4; ++mi) acc[mi] = wmma_f32(a[mi], bf, acc[mi]);
      bf = bfn;
#pragma unroll
      for (int mi = 0; mi < 4; ++mi) a[mi] = an[mi];
    }
#pragma unroll
    for (int mi = 0; mi < 4; ++mi) acc[mi] = wmma_f32(a[mi], bf, acc[mi]);

    __syncthreads();  // all waves finished reading h for this step

#pragma unroll
    for (int mi = 0; mi < 4; ++mi) {
#pragma unroll
      for (int r = 0; r < 8; ++r) {
        const int b  = mi * 16 + mb + r;
        const int sl = sseq[b];
        if (t < sl) {
          const float hn =
              fmaxf(acc[mi][r] + G[((size_t)t * BB + b) * HH + n], 0.0f);
          hs[b * HSTRIDE + n] = hn;                      // masked h update
          const int tt = dir ? (sl - 1 - t) : t;         // un-reverse for bwd
          outbuf[((size_t)b * TT + tt) * (2 * HH) + col0 + n] = hn;
        }
      }
    }
    __syncthreads();  // h writes visible before next step's reads
  }
}

// ---------------------------------------------------------------------------
// BatchNorm helpers: per-column stats over BT rows -> scale/shift, then
// in-place y = relu(x*scale + shift).
// ---------------------------------------------------------------------------
__global__ __launch_bounds__(256)
void bn_stats_kernel(const float* __restrict__ X, int ld, int rows,
                     const float* __restrict__ gamma, const float* __restrict__ beta,
                     float* __restrict__ scale, float* __restrict__ shift) {
  const int c = blockIdx.x;
  float s = 0.0f, sq = 0.0f;
  for (int r = threadIdx.x; r < rows; r += 256) {
    const float v = X[(size_t)r * ld + c];
    s += v;
    sq += v * v;
  }
  __shared__ float ss[256], ssq[256];
  ss[threadIdx.x]  = s;
  ssq[threadIdx.x] = sq;
  __syncthreads();
  for (int o = 128; o > 0; o >>= 1) {
    if (threadIdx.x < o) {
      ss[threadIdx.x]  += ss[threadIdx.x + o];
      ssq[threadIdx.x] += ssq[threadIdx.x + o];
    }
    __syncthreads();
  }
  if (threadIdx.x == 0) {
    const float mean = ss[0] / (float)rows;
    const float var  = ssq[0] / (float)rows - mean * mean;
    const float sc   = gamma[c] * rsqrtf(var + 1e-5f);
    scale[c] = sc;
    shift[c] = beta[c] - mean * sc;
  }
}

__global__ void bn_apply_relu_kernel(float* __restrict__ X,
                                     const float* __restrict__ sc,
                                     const float* __restrict__ sh,
                                     int ncols_mask, size_t total) {
  const size_t i = (size_t)blockIdx.x * blockDim.x + threadIdx.x;
  if (i < total) {
    const int c = (int)(i & (size_t)(ncols_mask));  // ncols is a power of two
    X[i] = fmaxf(X[i] * sc[c] + sh[c], 0.0f);
  }
}

// ---------------------------------------------------------------------------
// Small setup kernels
// ---------------------------------------------------------------------------
__global__ void extract_cluster_kernel(const float* __restrict__ x,
                                       int* __restrict__ cl) {
  const int b = threadIdx.x;
  if (b < BB) cl[b] = (int)x[(size_t)b * TT * (INF + 1)];
}

__global__ void rev_idx_kernel(const int* __restrict__ seq,
                               int* __restrict__ idx) {
  const int i = blockIdx.x * 256 + threadIdx.x;
  if (i < BT) {
    const int b = i >> 8, t = i & (TT - 1);
    int r = seq[b] - 1 - t;
    r = r < 0 ? 0 : (r > TT - 1 ? TT - 1 : r);
    idx[i] = b * TT + r;
  }
}

__global__ void combine_bias_kernel(const float* __restrict__ b1,
                                    const float* __restrict__ b2,
                                    float* __restrict__ o, int n) {
  const int i = blockIdx.x * 256 + threadIdx.x;
  if (i < n) o[i] = b1[i] + b2[i];
}

// ---------------------------------------------------------------------------
// Host orchestration
// ---------------------------------------------------------------------------
extern "C" void kernel_launch(void* const* d_in, const int* in_sizes, int n_in,
                              void* d_out, int out_size, void* d_ws,
                              size_t ws_size, hipStream_t stream) {
  (void)in_sizes; (void)n_in; (void)out_size; (void)ws_size;

  const float* x      = (const float*)d_in[0];
  const int*   seq    = (const int*)  d_in[1];
  const float* h0     = (const float*)d_in[2];
  const float* fc_w   = (const float*)d_in[3];
  const float* fc_b   = (const float*)d_in[4];
  const float* bn1_g  = (const float*)d_in[5];
  const float* bn1_b  = (const float*)d_in[6];
  const float* bn2_g  = (const float*)d_in[7];
  const float* bn2_b  = (const float*)d_in[8];
  const float* wih0f  = (const float*)d_in[9];
  const float* whh0f  = (const float*)d_in[10];
  const float* bih0f  = (const float*)d_in[11];
  const float* bhh0f  = (const float*)d_in[12];
  const float* wih0b  = (const float*)d_in[13];
  const float* whh0b  = (const float*)d_in[14];
  const float* bih0b  = (const float*)d_in[15];
  const float* bhh0b  = (const float*)d_in[16];
  const float* wih1f  = (const float*)d_in[17];
  const float* whh1f  = (const float*)d_in[18];
  const float* bih1f  = (const float*)d_in[19];
  const float* bhh1f  = (const float*)d_in[20];
  const float* wih1b  = (const float*)d_in[21];
  const float* whh1b  = (const float*)d_in[22];
  const float* bih1b  = (const float*)d_in[23];
  const float* bhh1b  = (const float*)d_in[24];
  const float* exp_w  = (const float*)d_in[25];
  const float* exp_b  = (const float*)d_in[26];
  float*       out    = (float*)d_out;

  // workspace layout (bytes)
  char* ws = (char*)d_ws;
  const size_t SZ_BT1024 = (size_t)BT * 1024 * 4;  // 64 MB
  const size_t SZ_BT512  = (size_t)BT * 512 * 4;   // 32 MB
  const size_t SZ_WHH    = (size_t)HH * HH * 4;    // 1 MB
  float* bufA  = (float*)(ws);                                  // 64 MB
  float* bufB  = (float*)(ws + SZ_BT1024);                      // 64 MB
  float* Gf    = (float*)(ws + 2 * SZ_BT1024);                  // 32 MB
  float* Gb    = (float*)(ws + 2 * SZ_BT1024 + SZ_BT512);       // 32 MB
  char*  tail  = ws + 2 * SZ_BT1024 + 2 * SZ_BT512;
  float* Wp0f  = (float*)(tail);               tail += SZ_WHH;  // packed Whh x4
  float* Wp0b  = (float*)(tail);               tail += SZ_WHH;
  float* Wp1f  = (float*)(tail);               tail += SZ_WHH;
  float* Wp1b  = (float*)(tail);               tail += SZ_WHH;
  int*   idx   = (int*)(tail);                 tail += (size_t)BT * 4;
  int*   cl    = (int*)(tail);                 tail += 256;
  float* b0f   = (float*)(tail);               tail += 512 * 4;
  float* b0b   = (float*)(tail);               tail += 512 * 4;
  float* b1f   = (float*)(tail);               tail += 512 * 4;
  float* b1b   = (float*)(tail);               tail += 512 * 4;
  float* scale = (float*)(tail);               tail += 1024 * 4;
  float* shift = (float*)(tail);

  // allow 132 KB dynamic LDS for the recurrence kernel
  (void)hipFuncSetAttribute(reinterpret_cast<const void*>(rnn_bidir_kernel),
                            hipFuncAttributeMaxDynamicSharedMemorySize,
                            RNN_LDS_BYTES);

  // setup
  extract_cluster_kernel<<<1, 64, 0, stream>>>(x, cl);
  rev_idx_kernel<<<BT / 256, 256, 0, stream>>>(seq, idx);
  combine_bias_kernel<<<2, 256, 0, stream>>>(bih0f, bhh0f, b0f, HH);
  combine_bias_kernel<<<2, 256, 0, stream>>>(bih0b, bhh0b, b0b, HH);
  combine_bias_kernel<<<2, 256, 0, stream>>>(bih1f, bhh1f, b1f, HH);
  combine_bias_kernel<<<2, 256, 0, stream>>>(bih1b, bhh1b, b1b, HH);
  pack_whh_kernel<<<512, 256, 0, stream>>>(whh0f, Wp0f);
  pack_whh_kernel<<<512, 256, 0, stream>>>(whh0b, Wp0b);
  pack_whh_kernel<<<512, 256, 0, stream>>>(whh1f, Wp1f);
  pack_whh_kernel<<<512, 256, 0, stream>>>(whh1b, Wp1b);

  // FC: bufB(BT x 512) = feats @ fc_w^T + fc_b   (feats = x[:,:,1:], lda=257)
  gemm_wmma_kernel<false><<<dim3(HH / 64, BT / 64, 1), 512, 0, stream>>>(
      x, INF + 1, 1, nullptr, fc_w, fc_b, nullptr, BT, HH, INF, bufB, HH, 0);
  // BN1 + ReLU (in place)
  bn_stats_kernel<<<HH, 256, 0, stream>>>(bufB, HH, BT, bn1_g, bn1_b, scale, shift);
  bn_apply_relu_kernel<<<(unsigned)((SZ_BT512 / 4 + 255) / 256), 256, 0, stream>>>(
      bufB, scale, shift, HH - 1, (size_t)BT * HH);

  // ---- layer 0 : gates stored (t,b,h) ----
  gemm_wmma_kernel<false><<<dim3(HH / 64, BT / 64, 1), 512, 0, stream>>>(
      bufB, HH, 0, nullptr, wih0f, b0f, nullptr, BT, HH, HH, Gf, HH, 1);
  gemm_wmma_kernel<false><<<dim3(HH / 64, BT / 64, 1), 512, 0, stream>>>(
      bufB, HH, 0, idx, wih0b, b0b, nullptr, BT, HH, HH, Gb, HH, 1);
  hipMemsetAsync(bufA, 0, SZ_BT1024, stream);
  rnn_bidir_kernel<<<2, 1024, RNN_LDS_BYTES, stream>>>(
      Gf, Gb, Wp0f, Wp0b, h0 + 0 * (size_t)BB * HH, seq, bufA);

  // ---- layer 1 (din = 1024) ----
  gemm_wmma_kernel<false><<<dim3(HH / 64, BT / 64, 1), 512, 0, stream>>>(
      bufA, 2 * HH, 0, nullptr, wih1f, b1f, nullptr, BT, HH, 2 * HH, Gf, HH, 1);
  gemm_wmma_kernel<false><<<dim3(HH / 64, BT / 64, 1), 512, 0, stream>>>(
      bufA, 2 * HH, 0, idx, wih1b, b1b, nullptr, BT, HH, 2 * HH, Gb, HH, 1);
  hipMemsetAsync(bufB, 0, SZ_BT1024, stream);
  rnn_bidir_kernel<<<2, 1024, RNN_LDS_BYTES, stream>>>(
      Gf, Gb, Wp1f, Wp1b, h0 + 2 * (size_t)BB * HH, seq, bufB);

  // BN2 + ReLU (in place on bufB, 1024 cols)
  bn_stats_kernel<<<2 * HH, 256, 0, stream>>>(bufB, 2 * HH, BT, bn2_g, bn2_b,
                                              scale, shift);
  bn_apply_relu_kernel<<<(unsigned)((SZ_BT1024 / 4 + 255) / 256), 256, 0, stream>>>(
      bufB, scale, shift, 2 * HH - 1, (size_t)BT * 2 * HH);

  // per-cluster expert head: out(BT x 256) = relu(bufB @ exp_w[cl[b]]^T + exp_b[cl[b]])
  gemm_wmma_kernel<true><<<dim3(OO / 64, TT / 64, BB), 512, 0, stream>>>(
      bufB, 2 * HH, 0, nullptr, exp_w, exp_b, cl, TT, OO, 2 * HH, out, OO, 0);
}